// FusionEmbedding_7851200217450
// MI455X (gfx1250) — compile-verified
//
#include <hip/hip_runtime.h>

// FusionEmbedding: out[t, :] = (id[t] < VOCAB) ? emb[id[t], :] : fusion[id[t]-VOCAB, :]
// Pure gather: HBM-bandwidth bound (~1 GB traffic -> ~44 us at 23.3 TB/s).
// Strategy: 1 block per token row, b128 loads, non-temporal b128 stores.

#define VOCAB_SIZE 128000
#define EMBED_DIM  4096   // 1024 float4 per row

typedef __attribute__((ext_vector_type(4))) float v4f;

__global__ __launch_bounds__(256) void fusion_embedding_kernel(
    const int*   __restrict__ ids,   // [ntok]
    const float* __restrict__ emb,   // [VOCAB_SIZE, EMBED_DIM]
    const float* __restrict__ fus,   // [ADDITIONAL, EMBED_DIM]
    float*       __restrict__ out,   // [ntok, EMBED_DIM]
    int ntok)
{
    const int tok = blockIdx.x;
    if (tok >= ntok) return;

    // Wave-uniform: compiler lowers this to s_load_b32.
    const int id = ids[tok];

    // Select source table (branchless on row pointer; uniform per block).
    const float* row = (id < VOCAB_SIZE)
        ? (emb + (size_t)id * EMBED_DIM)
        : (fus + (size_t)(id - VOCAB_SIZE) * EMBED_DIM);

    const v4f* __restrict__ src = (const v4f*)row;
    v4f*       __restrict__ dst = (v4f*)(out + (size_t)tok * EMBED_DIM);

    const int t = threadIdx.x;

    // 1024 float4 slots / 256 threads = 4 per thread.
    // Issue all 4 b128 loads first (4 outstanding LOADcnt entries per wave),
    // then drain with non-temporal b128 stores (write-once output: keep it
    // out of L2 so cached embedding rows survive for repeated token ids).
    v4f v0 = src[t];
    v4f v1 = src[t + 256];
    v4f v2 = src[t + 512];
    v4f v3 = src[t + 768];

    __builtin_nontemporal_store(v0, dst + t);
    __builtin_nontemporal_store(v1, dst + t + 256);
    __builtin_nontemporal_store(v2, dst + t + 512);
    __builtin_nontemporal_store(v3, dst + t + 768);
}

extern "C" void kernel_launch(void* const* d_in, const int* in_sizes, int n_in,
                              void* d_out, int out_size, void* d_ws, size_t ws_size,
                              hipStream_t stream)
{
    const int*   ids = (const int*)  d_in[0];  // [4, 8192] int32 token ids
    const float* emb = (const float*)d_in[1];  // [128000, 4096] fp32
    const float* fus = (const float*)d_in[2];  // [256, 4096] fp32
    float*       out = (float*)      d_out;    // [4, 8192, 4096] fp32

    const int ntok = in_sizes[0];              // 4 * 8192 = 32768

    fusion_embedding_kernel<<<dim3((unsigned)ntok), dim3(256), 0, stream>>>(
        ids, emb, fus, out, ntok);
}